// SelfAttention_v2_30855045054730
// MI455X (gfx1250) — compile-verified
//
#include <hip/hip_runtime.h>
#include <math.h>

typedef float v2f __attribute__((ext_vector_type(2)));
typedef float v8f __attribute__((ext_vector_type(8)));
typedef unsigned int u32x4 __attribute__((ext_vector_type(4)));
typedef int i32x4 __attribute__((ext_vector_type(4)));
typedef int i32x8 __attribute__((ext_vector_type(8)));

#define SEQ_LEN 8192
#define DIM_IN  768
#define DIM_OUT 64
#define BK      32              // keys per flash-attention block
#define KSTRIDE 66              // LDS row stride in floats (64 + 2 pad DWORDs)

// D = A(16x4 f32) * B(4x16 f32) + C(16x16 f32), full fp32 precision.
__device__ __forceinline__ v8f wmma_f32_16x16x4(v2f a, v2f b, v8f c) {
  return __builtin_amdgcn_wmma_f32_16x16x4_f32(
      /*neg_a=*/false, a, /*neg_b=*/false, b,
      /*c_mod=*/(short)0, c, /*reuse_a=*/false, /*reuse_b=*/false);
}

// ---------------------------------------------------------------------------
// TDM: async-load one 32x64 f32 tile (row stride 64 elements in memory) into
// LDS at lds_addr, padding +2 DWORDs after every 64 DWORDs (LDS stride 66).
// D# packing per CDNA5 ISA ch.8:
//   group0: [1:0]=count=1, [63:32]=lds_addr, [120:64]=global_addr, [127:126]=2
//   group1: [17:16]=data_size(2=4B), [20]=pad_en, [24:22]=pad_interval(5=64DW),
//           [31:25]=pad_amount(1=2DW), [79:48]=tensor_dim0=64,
//           [111:80]=tensor_dim1=8192, [127:112]=tile_dim0=64,
//           [143:128]=tile_dim1=32, [207:160]=tensor_dim0_stride=64
// ---------------------------------------------------------------------------
__device__ __forceinline__ void tdm_load_tile_32x64(const float* gsrc,
                                                    unsigned lds_addr) {
  const unsigned long long ga = (unsigned long long)(uintptr_t)gsrc;
  u32x4 g0;
  g0[0] = 1u;                                           // count=1
  g0[1] = lds_addr;                                     // LDS byte address
  g0[2] = (unsigned)ga;                                 // global_addr[31:0]
  g0[3] = (unsigned)((ga >> 32) & 0x1FFFFFFu) | (2u << 30);  // addr[56:32]|type=2
  i32x8 g1;
  g1[0] = (int)((2u << 16) | (1u << 20) | (5u << 22) | (1u << 25));
  g1[1] = (int)(64u << 16);        // tensor_dim0[15:0] = 64
  g1[2] = (int)(8192u << 16);      // tensor_dim0 hi=0 | tensor_dim1[15:0]=8192
  g1[3] = (int)(64u << 16);        // tensor_dim1 hi=0 | tile_dim0=64
  g1[4] = (int)32u;                // tile_dim1=32, tile_dim2=0
  g1[5] = (int)64u;                // tensor_dim0_stride = 64 elements
  g1[6] = 0;
  g1[7] = 0;
  const i32x4 z4 = {0, 0, 0, 0};             // groups 2/3 unused (2-D tensor)
  const i32x8 z8 = {0, 0, 0, 0, 0, 0, 0, 0}; // trailing optional group unused
  __builtin_amdgcn_tensor_load_to_lds(g0, g1, z4, z4, z8, 0);
}

// ---------------------------------------------------------------------------
// Kernel 1: q/k/v = x @ W_{q,k,v}.  grid = (SEQ/16, 3), block = 32 (1 wave).
// Fragment layouts (ISA 7.12.2, 32-bit):
//   A 16x4 (MxK): m=lane&15, g=lane>>4; vgpr j holds A[m][kbase+2g+j]
//   B 4x16 (KxN): n=lane&15, g=lane>>4; vgpr j holds B[kbase+2g+j][n]
//   C 16x16:      n=lane&15, g=lane>>4; vgpr i holds C[8g+i][n]
// ---------------------------------------------------------------------------
__global__ __launch_bounds__(32) void qkv_proj_f32wmma(
    const float* __restrict__ x,
    const float* __restrict__ Wq,
    const float* __restrict__ Wk,
    const float* __restrict__ Wv,
    float* __restrict__ qkv) {
  const int lane = threadIdx.x & 31;
  const int m    = lane & 15;
  const int g    = lane >> 4;

  const int which = blockIdx.y;  // 0=q, 1=k, 2=v
  const float* __restrict__ W =
      (which == 0) ? Wq : (which == 1) ? Wk : Wv;
  float* __restrict__ outp = qkv + (size_t)which * SEQ_LEN * DIM_OUT;

  const int mbase = blockIdx.x * 16;
  const float* __restrict__ xrow = x + (size_t)(mbase + m) * DIM_IN;

  const v8f zero8 = {0.f, 0.f, 0.f, 0.f, 0.f, 0.f, 0.f, 0.f};
  v8f acc[4];
#pragma unroll
  for (int t = 0; t < 4; ++t) acc[t] = zero8;

#pragma unroll 4
  for (int k0 = 0; k0 < DIM_IN; k0 += 4) {
    const v2f a = *(const v2f*)(xrow + k0 + 2 * g);
#pragma unroll
    for (int t = 0; t < 4; ++t) {
      v2f b;
      b.x = W[(size_t)(k0 + 2 * g + 0) * DIM_OUT + t * 16 + m];
      b.y = W[(size_t)(k0 + 2 * g + 1) * DIM_OUT + t * 16 + m];
      acc[t] = wmma_f32_16x16x4(a, b, acc[t]);
    }
  }

#pragma unroll
  for (int t = 0; t < 4; ++t)
#pragma unroll
    for (int i = 0; i < 8; ++i)
      outp[(size_t)(mbase + 8 * g + i) * DIM_OUT + t * 16 + m] = acc[t][i];
}

// ---------------------------------------------------------------------------
// Kernel 2: flash attention with TDM double-buffered K/V staging.
// grid = SEQ/16 blocks, block = 32 threads (1 wave per 16-row query strip).
// Processes keys in blocks of 32 (two 16x16 S tiles per iteration).
// ---------------------------------------------------------------------------
__global__ __launch_bounds__(32) void flash_attn_f32wmma(
    const float* __restrict__ qkv, float* __restrict__ out) {
  const float* __restrict__ qm = qkv;
  const float* __restrict__ km = qkv + (size_t)1 * SEQ_LEN * DIM_OUT;
  const float* __restrict__ vm = qkv + (size_t)2 * SEQ_LEN * DIM_OUT;

  // [buffer][0=K,1=V][32 rows * 66 floats] padded tiles written by the TDM.
  __shared__ float kv[2][2][BK * KSTRIDE];
  // P tile staging (C-layout -> A-layout); stride 34 keeps b64 reads aligned.
  __shared__ float pbuf[16 * 34];

  const int lane  = threadIdx.x & 31;
  const int m     = lane & 15;
  const int g     = lane >> 4;
  const int mbase = blockIdx.x * 16;

  // Preload q strip as 16 pre-scaled A-fragments (K=64 -> 16 steps of 4).
  v2f aq[16];
  const float* __restrict__ qrow = qm + (size_t)(mbase + m) * DIM_OUT;
#pragma unroll
  for (int kk = 0; kk < 16; ++kk) {
    v2f t = *(const v2f*)(qrow + 4 * kk + 2 * g);
    t.x *= 0.125f;
    t.y *= 0.125f;
    aq[kk] = t;
  }

  const v8f zero8 = {0.f, 0.f, 0.f, 0.f, 0.f, 0.f, 0.f, 0.f};
  v8f acc[4];
#pragma unroll
  for (int t = 0; t < 4; ++t) acc[t] = zero8;

  float mrun[8], lrun[8];
#pragma unroll
  for (int i = 0; i < 8; ++i) {
    mrun[i] = -INFINITY;
    lrun[i] = 0.f;
  }

  // Prefetch first K/V pair into buffer 0.
  tdm_load_tile_32x64(km, (unsigned)(uintptr_t)&kv[0][0][0]);
  tdm_load_tile_32x64(vm, (unsigned)(uintptr_t)&kv[0][1][0]);

  for (int j = 0; j < SEQ_LEN; j += BK) {
    const int buf = (j >> 5) & 1;

    if (j + BK < SEQ_LEN) {
      // Issue next pair into the other buffer, then wait for current pair
      // (TDM ops of one wave complete in order: <=2 outstanding == done).
      const int nb = buf ^ 1;
      tdm_load_tile_32x64(km + (size_t)(j + BK) * DIM_OUT,
                          (unsigned)(uintptr_t)&kv[nb][0][0]);
      tdm_load_tile_32x64(vm + (size_t)(j + BK) * DIM_OUT,
                          (unsigned)(uintptr_t)&kv[nb][1][0]);
      __builtin_amdgcn_s_wait_tensorcnt(2);
    } else {
      __builtin_amdgcn_s_wait_tensorcnt(0);
    }

    const float* __restrict__ kt = &kv[buf][0][0];
    const float* __restrict__ vt = &kv[buf][1][0];

    // ---- S tiles: s[nt] = (q*0.125) @ K[j+16nt .. +15]^T ----
    v8f s0 = zero8, s1 = zero8;
#pragma unroll
    for (int kk = 0; kk < 16; ++kk) {
      const v2f b0 = *(const v2f*)(kt + (0 * 16 + m) * KSTRIDE + 4 * kk + 2 * g);
      const v2f b1 = *(const v2f*)(kt + (1 * 16 + m) * KSTRIDE + 4 * kk + 2 * g);
      s0 = wmma_f32_16x16x4(aq[kk], b0, s0);
      s1 = wmma_f32_16x16x4(aq[kk], b1, s1);
    }

    // ---- online softmax over this 32-key block ----
    float alpha[8];
#pragma unroll
    for (int i = 0; i < 8; ++i) {
      float rm = fmaxf(s0[i], s1[i]);     // combine both tiles first
      rm = fmaxf(rm, __shfl_xor(rm, 1, 32));
      rm = fmaxf(rm, __shfl_xor(rm, 2, 32));
      rm = fmaxf(rm, __shfl_xor(rm, 4, 32));
      rm = fmaxf(rm, __shfl_xor(rm, 8, 32));
      const float mnew = fmaxf(mrun[i], rm);
      const float p0   = __expf(s0[i] - mnew);
      const float p1   = __expf(s1[i] - mnew);
      float rs = p0 + p1;
      rs += __shfl_xor(rs, 1, 32);
      rs += __shfl_xor(rs, 2, 32);
      rs += __shfl_xor(rs, 4, 32);
      rs += __shfl_xor(rs, 8, 32);
      alpha[i] = __expf(mrun[i] - mnew);
      lrun[i]  = lrun[i] * alpha[i] + rs;
      mrun[i]  = mnew;
      // scatter P in C-layout: row = 8g+i, cols nt*16+m
      pbuf[(8 * g + i) * 34 + 0 * 16 + m] = p0;
      pbuf[(8 * g + i) * 34 + 1 * 16 + m] = p1;
    }
    __syncthreads();  // single-wave WG: orders LDS store->load

    // rescale running accumulator
#pragma unroll
    for (int t = 0; t < 4; ++t)
#pragma unroll
      for (int i = 0; i < 8; ++i) acc[t][i] *= alpha[i];

    // ---- acc += P @ V  (contraction over 32 keys -> 8 WMMA steps) ----
#pragma unroll
    for (int kb = 0; kb < BK; kb += 4) {
      const v2f ap = *(const v2f*)(&pbuf[m * 34 + kb + 2 * g]);
#pragma unroll
      for (int t = 0; t < 4; ++t) {
        v2f bv;
        bv.x = vt[(kb + 2 * g + 0) * KSTRIDE + t * 16 + m];
        bv.y = vt[(kb + 2 * g + 1) * KSTRIDE + t * 16 + m];
        acc[t] = wmma_f32_16x16x4(ap, bv, acc[t]);
      }
    }
    __syncthreads();  // protect pbuf before next key block overwrites it
  }

  // ---- epilogue: out = acc / l ----
  float inv[8];
#pragma unroll
  for (int i = 0; i < 8; ++i) inv[i] = 1.0f / lrun[i];
#pragma unroll
  for (int t = 0; t < 4; ++t)
#pragma unroll
    for (int i = 0; i < 8; ++i)
      out[(size_t)(mbase + 8 * g + i) * DIM_OUT + t * 16 + m] =
          acc[t][i] * inv[i];
}

extern "C" void kernel_launch(void* const* d_in, const int* in_sizes, int n_in,
                              void* d_out, int out_size, void* d_ws,
                              size_t ws_size, hipStream_t stream) {
  const float* x  = (const float*)d_in[0];
  const float* Wq = (const float*)d_in[1];
  const float* Wk = (const float*)d_in[2];
  const float* Wv = (const float*)d_in[3];
  float* out = (float*)d_out;
  float* qkv = (float*)d_ws;

  const size_t need = (size_t)3 * SEQ_LEN * DIM_OUT * sizeof(float);
  if (ws_size < need) return;  // workspace too small; nothing safe to do

  qkv_proj_f32wmma<<<dim3(SEQ_LEN / 16, 3), 32, 0, stream>>>(x, Wq, Wk, Wv,
                                                             qkv);
  flash_attn_f32wmma<<<dim3(SEQ_LEN / 16), 32, 0, stream>>>(qkv, out);
}